// MultiViewMoE_3453153706431
// MI455X (gfx1250) — compile-verified
//
#include <hip/hip_runtime.h>

typedef __attribute__((ext_vector_type(2))) float v2f;
typedef __attribute__((ext_vector_type(4))) float v4f;
typedef __attribute__((ext_vector_type(8))) float v8f;
typedef int v4i __attribute__((vector_size(16)));

#define N_NODES   16384
#define N_EDGES   65536
#define DIM       256
#define EDIM      24
#define GHID      2048
#define NGRAPH    64
#define NEXP      8
#define NVIEW     3

#define FL_RELU 1
#define FL_ACC  2
#define FL_GATE 4

// ---------------- WMMA helpers (fp32 16x16x4) ----------------
__device__ inline v8f wmma_f32(v2f a, v2f b, v8f c) {
  return __builtin_amdgcn_wmma_f32_16x16x4_f32(false, a, false, b, (short)0, c, false, false);
}

// A-frag 16x4 (MxK): lane<16 -> row=lane, {k0,k0+1}; lane>=16 -> row=lane-16, {k0+2,k0+3}
__device__ inline v2f ldA(const float* As, int lane, int k0, int lda) {
  int m = lane & 15;
  int k = k0 + ((lane >> 4) << 1);
  v2f a; a.x = As[m * lda + k]; a.y = As[m * lda + k + 1]; return a;
}
// B-frag 4x16 (KxN): lane<16 -> col=lane rows {k0,k0+1}; lane>=16 -> col=lane-16 rows {k0+2,k0+3}
__device__ inline v2f ldB(const float* Bs, int lane, int k0, int ldb) {
  int n = lane & 15;
  int k = k0 + ((lane >> 4) << 1);
  v2f b; b.x = Bs[k * ldb + n]; b.y = Bs[(k + 1) * ldb + n]; return b;
}

// ---------------- Async global->LDS 16-byte copy (CDNA5 GLOBAL_LOAD_ASYNC_TO_LDS_B128) ----
__device__ inline void async_cp16(const float* g, float* lds) {
#if __has_builtin(__builtin_amdgcn_global_load_async_to_lds_b128)
  __builtin_amdgcn_global_load_async_to_lds_b128(
      (__attribute__((address_space(1))) v4i*)(void*)(g),
      (__attribute__((address_space(3))) v4i*)(void*)(lds), 0, 0);
#else
  *(v4f*)lds = *(const v4f*)g;
#endif
}
__device__ inline void wait_async() {
#if __has_builtin(__builtin_amdgcn_s_wait_asynccnt)
  __builtin_amdgcn_s_wait_asynccnt(0);
#else
#if __has_builtin(__builtin_amdgcn_global_load_async_to_lds_b128)
  asm volatile("s_wait_asynccnt 0x0" ::: "memory");
#endif
#endif
}

// ---------------- Generic WMMA GEMM: C = epi(A[M,K] @ W[K,N] + bias) ----------------
// grid = (M/64, N/64), 256 threads (8 waves); wave w: M-subtile w&3, N-subtiles 2*(w>>2)+{0,1}
// Double-buffered K loop: async tile copy for step k+1 overlaps WMMA on step k.
template <int FLAGS>
__global__ __launch_bounds__(256)
void k_gemm(const float* __restrict__ A, int lda,
            const float* __restrict__ W, int ldw,
            const float* __restrict__ bias,
            float* __restrict__ C, int ldc,
            int K, const float* __restrict__ gate, int gstride)
{
  __shared__ __align__(16) float As[2][64 * 16];
  __shared__ __align__(16) float Bs[2][16 * 64];
  const int tid = threadIdx.x;
  const int wid = tid >> 5, lane = tid & 31;
  const int mb = blockIdx.x * 64, nb = blockIdx.y * 64;
  const int msub = wid & 3;
  const int nsub0 = (wid >> 2) * 2;
  // per-thread 16B chunk indices (1024 floats per tile / 256 threads)
  const int ia = tid * 4, ar = ia >> 4, ac = ia & 15;
  const int br = ia >> 6, bc = ia & 63;

  v8f acc0 = {}, acc1 = {};
  // prologue: stage tile 0
  async_cp16(&A[(size_t)(mb + ar) * lda + ac], &As[0][ia]);
  async_cp16(&W[(size_t)br * ldw + nb + bc], &Bs[0][ia]);

  int buf = 0;
  for (int k0 = 0; k0 < K; k0 += 16) {
    wait_async();
    __syncthreads();
    if (k0 + 16 < K) {  // prefetch next tile into the other buffer
      async_cp16(&A[(size_t)(mb + ar) * lda + (k0 + 16) + ac], &As[buf ^ 1][ia]);
      async_cp16(&W[(size_t)(k0 + 16 + br) * ldw + nb + bc], &Bs[buf ^ 1][ia]);
    }
    const float* Asub = As[buf] + msub * 16 * 16;
    const float* B0 = Bs[buf] + nsub0 * 16;
    const float* B1 = B0 + 16;
    #pragma unroll
    for (int kk = 0; kk < 16; kk += 4) {
      v2f a  = ldA(Asub, lane, kk, 16);
      v2f b0 = ldB(B0, lane, kk, 64);
      v2f b1 = ldB(B1, lane, kk, 64);
      acc0 = wmma_f32(a, b0, acc0);
      acc1 = wmma_f32(a, b1, acc1);
    }
    buf ^= 1;
  }
  const int lhalf = (lane >> 4) * 8;
  const int lcol = lane & 15;
  #pragma unroll
  for (int j = 0; j < 2; ++j) {
    v8f acc = j ? acc1 : acc0;
    int colg = nb + (nsub0 + j) * 16 + lcol;
    float bv = bias ? bias[colg] : 0.f;
    #pragma unroll
    for (int r = 0; r < 8; ++r) {
      int rowg = mb + msub * 16 + r + lhalf;
      float v = acc[r] + bv;
      if (FLAGS & FL_RELU) v = fmaxf(v, 0.f);
      if (FLAGS & FL_GATE) v *= gate[(size_t)rowg * gstride];
      float* cp = &C[(size_t)rowg * ldc + colg];
      if (FLAGS & FL_ACC) *cp += v; else *cp = v;
    }
  }
}

// ---------------- Fused edge kernel: emsg WMMA (K=24) + gather h[src] + relu + scatter-add ----
// grid = (E/64, 256/64)
__global__ __launch_bounds__(256)
void k_edge(const float* __restrict__ ea, const float* __restrict__ We,
            const float* __restrict__ be, const float* __restrict__ h,
            const int* __restrict__ src, const int* __restrict__ dst,
            float* __restrict__ agg)
{
  __shared__ __align__(16) float Ea[64 * 24];
  __shared__ __align__(16) float Ws[24 * 64];
  const int tid = threadIdx.x;
  const int wid = tid >> 5, lane = tid & 31;
  const int eb = blockIdx.x * 64;
  const int nb = blockIdx.y * 64;
  // 64x24 tile = 384 16B-chunks; 24%4==0 so chunks never straddle a row
  for (int t = tid; t < 384; t += 256) {
    int i4 = t * 4;
    int r = i4 / 24, c = i4 - r * 24;
    async_cp16(&ea[(size_t)(eb + r) * EDIM + c], &Ea[i4]);
  }
  for (int t = tid; t < 384; t += 256) {
    int i4 = t * 4;
    int r = i4 >> 6, c = i4 & 63;
    async_cp16(&We[(size_t)r * DIM + nb + c], &Ws[i4]);
  }
  wait_async();
  __syncthreads();
  const int msub = wid & 3;
  const int nsub0 = (wid >> 2) * 2;
  v8f acc0 = {}, acc1 = {};
  const float* Asub = Ea + msub * 16 * 24;
  const float* B0 = Ws + nsub0 * 16;
  const float* B1 = B0 + 16;
  #pragma unroll
  for (int kk = 0; kk < 24; kk += 4) {
    v2f a  = ldA(Asub, lane, kk, 24);
    v2f b0 = ldB(B0, lane, kk, 64);
    v2f b1 = ldB(B1, lane, kk, 64);
    acc0 = wmma_f32(a, b0, acc0);
    acc1 = wmma_f32(a, b1, acc1);
  }
  const int lhalf = (lane >> 4) * 8;
  const int lcol = lane & 15;
  #pragma unroll
  for (int j = 0; j < 2; ++j) {
    v8f acc = j ? acc1 : acc0;
    int colg = nb + (nsub0 + j) * 16 + lcol;
    float bv = be[colg];
    #pragma unroll
    for (int r = 0; r < 8; ++r) {
      int eg = eb + msub * 16 + r + lhalf;
      float emsg = fmaxf(acc[r] + bv, 0.f);
      int s = src[eg], d = dst[eg];
      float m = fmaxf(h[(size_t)s * DIM + colg] + emsg, 0.f);
      atomicAdd(&agg[(size_t)d * DIM + colg], m);
    }
  }
}

// ---------------- Small kernels ----------------
__global__ void k_zero(float* __restrict__ p, int n) {
  int i = blockIdx.x * blockDim.x + threadIdx.x;
  if (i < n) p[i] = 0.f;
}

// one block per node, DIM threads
__global__ void k_seg_accum(const float* __restrict__ x, const int* __restrict__ batch,
                            float* __restrict__ sum, float* __restrict__ cnt)
{
  int n = blockIdx.x;
  int g = batch[n];
  int d = threadIdx.x;
  atomicAdd(&sum[(size_t)g * DIM + d], x[(size_t)n * DIM + d]);
  if (cnt && d == 0) atomicAdd(&cnt[g], 1.f);
}

__global__ void k_div(const float* __restrict__ sum, const float* __restrict__ cnt,
                      float* __restrict__ outp) {
  int g = blockIdx.x, d = threadIdx.x;
  outp[(size_t)g * DIM + d] = sum[(size_t)g * DIM + d] / fmaxf(cnt[g], 1.f);
}

// one block per graph: logits[g,e] = dot(gfh[g,:], Wg[:,e]) + bg[e]
__global__ void k_logits(const float* __restrict__ gfh, const float* __restrict__ Wg,
                         const float* __restrict__ bg, float* __restrict__ logits)
{
  __shared__ float red[256];
  int g = blockIdx.x, t = threadIdx.x;
  for (int e = 0; e < NEXP; ++e) {
    float p = 0.f;
    for (int i = t; i < GHID; i += 256) p += gfh[(size_t)g * GHID + i] * Wg[(size_t)i * NEXP + e];
    red[t] = p; __syncthreads();
    for (int s = 128; s > 0; s >>= 1) { if (t < s) red[t] += red[t + s]; __syncthreads(); }
    if (t == 0) logits[g * NEXP + e] = red[0] + bg[e];
    __syncthreads();
  }
}

// one thread per graph: top-4, softmax, dense gates, importance, L2 norm
__global__ void k_topk(const float* __restrict__ logits, float* __restrict__ gates,
                       float* __restrict__ imp, float* __restrict__ norms)
{
  int g = threadIdx.x;
  if (g >= NGRAPH) return;
  float l[NEXP]; float nrm = 0.f;
  for (int e = 0; e < NEXP; ++e) {
    l[e] = logits[g * NEXP + e];
    nrm += l[e] * l[e];
    gates[g * NEXP + e] = 0.f;
  }
  norms[g] = sqrtf(nrm);
  int idx[4]; float val[4]; bool used[NEXP] = {};
  for (int k = 0; k < 4; ++k) {
    int bi = 0; float bv = -3.4e38f;
    for (int e = 0; e < NEXP; ++e)
      if (!used[e] && l[e] > bv) { bv = l[e]; bi = e; }
    used[bi] = true; idx[k] = bi; val[k] = bv;
  }
  float mx = val[0], s = 0.f, ex[4];
  for (int k = 0; k < 4; ++k) { ex[k] = expf(val[k] - mx); s += ex[k]; }
  for (int k = 0; k < 4; ++k) {
    gates[g * NEXP + idx[k]] = ex[k] / s;
    atomicAdd(&imp[idx[k]], 1.f);
  }
}

__global__ void k_subw(const float* __restrict__ norms, float* __restrict__ subw) {
  __shared__ float mn, mx;
  int g = threadIdx.x;
  if (g == 0) {
    float a = norms[0], b = norms[0];
    for (int i = 1; i < NGRAPH; ++i) { a = fminf(a, norms[i]); b = fmaxf(b, norms[i]); }
    mn = a; mx = b;
  }
  __syncthreads();
  if (g < NGRAPH) subw[g] = (norms[g] - mn) / (mx - mn + 1e-6f) + 0.1f;
}

__global__ void k_node_gates(const int* __restrict__ batch, const float* __restrict__ gates,
                             float* __restrict__ ng)
{
  int i = blockIdx.x * blockDim.x + threadIdx.x;
  if (i >= N_NODES * NEXP) return;
  int n = i >> 3, e = i & 7;
  ng[i] = gates[batch[n] * NEXP + e];
}

__global__ void k_loss(const float* __restrict__ imp, float* __restrict__ outp) {
  if (threadIdx.x || blockIdx.x) return;
  float m = 0.f;
  for (int e = 0; e < NEXP; ++e) m += imp[e];
  m /= (float)NEXP;
  float v = 0.f;
  for (int e = 0; e < NEXP; ++e) { float d = imp[e] - m; v += d * d; }
  v /= (float)(NEXP - 1);                 // ddof=1
  float cv = v / (m * m + 1e-10f);
  outp[0] = (cv + cv) * 0.01f;
}

// ---------------- Launch ----------------
extern "C" void kernel_launch(void* const* d_in, const int* in_sizes, int n_in,
                              void* d_out, int out_size, void* d_ws, size_t ws_size,
                              hipStream_t stream)
{
  (void)in_sizes; (void)n_in; (void)out_size; (void)ws_size;
  const float* xv    = (const float*)d_in[0];
  const int*   eiv   = (const int*)d_in[1];
  const int*   batch = (const int*)d_in[2];
  const float* eav   = (const float*)d_in[3];
  const float* Wg1   = (const float*)d_in[5];
  const float* bg1   = (const float*)d_in[6];
  const float* Wg    = (const float*)d_in[7];
  const float* bg    = (const float*)d_in[8];
  const float* W_in  = (const float*)d_in[11];
  const float* b_in  = (const float*)d_in[12];
  const float* W_e   = (const float*)d_in[13];
  const float* b_e   = (const float*)d_in[14];
  const float* W_out = (const float*)d_in[15];
  const float* b_out = (const float*)d_in[16];
  float* outp = (float*)d_out;
  float* ws   = (float*)d_ws;

  // workspace layout (floats)
  float* gf_sum = ws;                 // 16384 (+64 cnt, zeroed together)
  float* cnt    = ws + 16384;         // 64
  float* gf     = ws + 16448;         // 16384
  float* logits = ws + 32832;         // 512
  float* gates  = ws + 33344;         // 512
  float* norms  = ws + 33856;         // 64
  float* imp    = ws + 33920;         // 8
  float* ng     = ws + 33928;         // 131072
  float* h_e    = ws + 165000;        // 4194304 (16 MB, L2-resident)
  float* agg_e  = ws + 4359304;       // 4194304 (16 MB, L2-resident)

  // output layout (flattened return order)
  const size_t FN = (size_t)N_NODES * DIM;             // 4194304 per view
  float* xg_base = outp + 3 * FN;                      // [3,64,256]
  float* loss_p  = xg_base + 3 * NGRAPH * DIM;         // scalar
  float* subcls0 = loss_p + 1;                         // 3 x [64,2048]
  float* subw0   = subcls0 + (size_t)3 * NGRAPH * GHID;// 3 x [64]

  k_zero<<<1, 32, 0, stream>>>(imp, NEXP);

  for (int v = 0; v < NVIEW; ++v) {
    const float* x_v  = xv  + (size_t)v * N_NODES * DIM;
    const float* ea_v = eav + (size_t)v * N_EDGES * EDIM;
    const int* src = eiv + (size_t)v * 2 * N_EDGES;
    const int* dst = src + N_EDGES;
    float* fn_v     = outp + (size_t)v * FN;
    float* xg_v     = xg_base + v * NGRAPH * DIM;
    float* subcls_v = subcls0 + (size_t)v * NGRAPH * GHID;
    float* subw_v   = subw0 + v * NGRAPH;

    // ---- gating ----
    k_zero<<<(16448 + 255) / 256, 256, 0, stream>>>(gf_sum, 16448);
    k_seg_accum<<<N_NODES, DIM, 0, stream>>>(x_v, batch, gf_sum, cnt);
    k_div<<<NGRAPH, DIM, 0, stream>>>(gf_sum, cnt, gf);
    // gfh = relu(gf@Wg1 + bg1) -> subcls output (also input to logits)
    k_gemm<FL_RELU><<<dim3(1, GHID / 64), 256, 0, stream>>>(
        gf, DIM, Wg1, GHID, bg1, subcls_v, GHID, DIM, nullptr, 0);
    k_logits<<<NGRAPH, 256, 0, stream>>>(subcls_v, Wg, bg, logits);
    k_topk<<<1, NGRAPH, 0, stream>>>(logits, gates, imp, norms);
    k_subw<<<1, NGRAPH, 0, stream>>>(norms, subw_v);
    k_node_gates<<<(N_NODES * NEXP + 255) / 256, 256, 0, stream>>>(batch, gates, ng);

    // ---- per-expert streamed EdgeGCN (all buffers L2-resident) ----
    k_zero<<<(int)((FN + 255) / 256), 256, 0, stream>>>(fn_v, (int)FN);
    for (int e = 0; e < NEXP; ++e) {
      // h_e = x_v @ W_in[e] + b_in[e]
      k_gemm<0><<<dim3(N_NODES / 64, DIM / 64), 256, 0, stream>>>(
          x_v, DIM, W_in + (size_t)e * DIM * DIM, DIM, b_in + (size_t)e * DIM,
          h_e, DIM, DIM, nullptr, 0);
      k_zero<<<(int)((FN + 255) / 256), 256, 0, stream>>>(agg_e, (int)FN);
      // agg_e[dst] += relu(h_e[src] + relu(ea@W_e[e] + b_e[e]))
      k_edge<<<dim3(N_EDGES / 64, DIM / 64), 256, 0, stream>>>(
          ea_v, W_e + (size_t)e * EDIM * DIM, b_e + (size_t)e * DIM, h_e, src, dst, agg_e);
      // fn_v += ng[:,e] * (agg_e @ W_out[e] + b_out[e])
      k_gemm<FL_ACC | FL_GATE><<<dim3(N_NODES / 64, DIM / 64), 256, 0, stream>>>(
          agg_e, DIM, W_out + (size_t)e * DIM * DIM, DIM, b_out + (size_t)e * DIM,
          fn_v, DIM, DIM, ng + e, NEXP);
    }

    // ---- graph-level mean of view output ----
    k_zero<<<(NGRAPH * DIM + 255) / 256, 256, 0, stream>>>(xg_v, NGRAPH * DIM);
    k_seg_accum<<<N_NODES, DIM, 0, stream>>>(fn_v, batch, xg_v, nullptr);
    k_div<<<NGRAPH, DIM, 0, stream>>>(xg_v, cnt, xg_v);
  }
  k_loss<<<1, 1, 0, stream>>>(imp, loss_p);
}